// FieldWeightedFactorizationMachine_60309930770650
// MI455X (gfx1250) — compile-verified
//
#include <hip/hip_runtime.h>

// FwFM forward for MI455X (gfx1250, wave32).
// output[b] = w0 + sum_{f,d} cfe[b,f,d] * (field_emb[f,d] + (Ueff @ cfe[b])[f,d])
// with Ueff = 0.25*(W+W^T), zero diagonal (folds symmetrization + trace removal).
// Core GEMM (40x40 @ 40x128 per batch element) runs on V_WMMA_F32_16X16X4_F32.
// The per-b row gather is software-pipelined: next b's rows are prefetched into
// registers while the current b's WMMAs run, hiding global-load latency.

#define B_SZ 8192
#define F_SZ 40
#define V_SZ 10000
#define D_SZ 128
#define FP   48          // F padded to 3 WMMA M-tiles
#define MT   3           // M tiles (48/16)
#define KT   10          // K steps  (40/4)
#define LS   132         // LDS row stride in floats (16B-aligned rows, bank-skewed)
#define GRID 2048
#define ITERS (B_SZ / GRID)   // 4, uniform across all blocks

typedef __attribute__((ext_vector_type(2))) float v2f;
typedef __attribute__((ext_vector_type(4))) float v4f;
typedef __attribute__((ext_vector_type(8))) float v8f;

// ---------------------------------------------------------------------------
// Kernel 1: pack Ueff (48x40, zero-padded, zero diagonal) into WMMA A-fragment
// layout: ws_a[t*64 + part*32 + lane], t = mt*KT + kt.
//   A 16x4 f32 tile: VGPR0 (part 0): K = 0 (lanes 0-15) / 2 (lanes 16-31)
//                    VGPR1 (part 1): K = 1 (lanes 0-15) / 3 (lanes 16-31)
//   M = mt*16 + (lane & 15)
// ---------------------------------------------------------------------------
__global__ void fwfm_pack_ueff(const float* __restrict__ W,
                               float* __restrict__ ws_a) {
    int i = blockIdx.x * blockDim.x + threadIdx.x;
    if (i >= MT * KT * 64) return;
    int lane = i & 31;
    int part = (i >> 5) & 1;
    int t    = i >> 6;
    int mt = t / KT, kt = t % KT;
    int m = mt * 16 + (lane & 15);
    int k = kt * 4 + part + ((lane < 16) ? 0 : 2);
    float v = 0.0f;
    if (m < F_SZ && m != k)   // k < 40 always (KT*4 == 40)
        v = 0.25f * (W[m * F_SZ + k] + W[k * F_SZ + m]);
    ws_a[i] = v;
}

// ---------------------------------------------------------------------------
// Kernel 2: main fused kernel. 256 threads = 8 waves; one batch element per
// block-iteration, ITERS uniform iterations (EXEC stays all-ones for WMMA).
// ---------------------------------------------------------------------------
__global__ __launch_bounds__(256)
void fwfm_main(const int*   __restrict__ index,      // (B,F)
               const float* __restrict__ coef,       // (B,F)
               const float* __restrict__ emb,        // (F,V,D)
               const float* __restrict__ field_emb,  // (F,D)
               const float* __restrict__ w0,         // (1,)
               const float* __restrict__ ws_a,       // packed A fragments
               float*       __restrict__ out_scalar, // (B,)
               float*       __restrict__ out_cfe)    // (B,F,D)
{
    __shared__ float s_cfe[FP * LS];   // cfe rows (rows 40..47 stay zero)
    __shared__ float s_fe [FP * LS];   // field_emb rows (rows 40..47 zero)
    __shared__ float s_red[8];

    const int tid  = threadIdx.x;
    const int wave = tid >> 5;
    const int lane = tid & 31;
    const int half = (lane < 16) ? 0 : 1;
    const int l16  = lane & 15;
    const int dbase = wave * 16;       // each wave owns a 16-wide d chunk

    // --- A fragments for all 30 tiles, resident in VGPRs for the whole block.
    v2f afrag[MT * KT];
#pragma unroll
    for (int t = 0; t < MT * KT; ++t) {
        afrag[t].x = ws_a[t * 64 + lane];
        afrag[t].y = ws_a[t * 64 + 32 + lane];
    }

    // --- Stage field_emb into LDS (padded stride); zero the pad rows.
    for (int i = tid; i < F_SZ * (D_SZ / 4); i += 256) {
        int f  = i / (D_SZ / 4);
        int d4 = (i % (D_SZ / 4)) * 4;
        v4f fe4 = *(const v4f*)(field_emb + f * D_SZ + d4);
        *(v4f*)(&s_fe[f * LS + d4]) = fe4;
    }
    {   // pad rows 40..47 of both LDS tiles (one per wave), written once
        int f = F_SZ + wave;
        for (int d = lane; d < LS; d += 32) {
            s_fe [f * LS + d] = 0.0f;
            s_cfe[f * LS + d] = 0.0f;
        }
    }

    const float w0v = w0[0];

    // --- Prologue: prefetch first b's rows (raw) + coefs into registers.
    v4f   pre [5];
    float cpre[5];
    {
        int b0 = blockIdx.x;
#pragma unroll
        for (int r = 0; r < 5; ++r) {
            int f = wave + r * 8;
            cpre[r] = coef[b0 * F_SZ + f];
            int ix  = index[b0 * F_SZ + f];
            pre[r]  = *(const v4f*)(emb + ((size_t)f * V_SZ + (size_t)ix) * D_SZ
                                        + lane * 4);
        }
    }

    for (int iter = 0; iter < ITERS; ++iter) {
        const int b = blockIdx.x + iter * GRID;
        __syncthreads();   // prior iteration's LDS readers done; init visible

        // ---- Commit prefetched rows: scale by coef, store to LDS + global.
#pragma unroll
        for (int r = 0; r < 5; ++r) {
            int f = wave + r * 8;
            v4f v = pre[r];
            float c = cpre[r];
            v.x *= c; v.y *= c; v.z *= c; v.w *= c;
            *(v4f*)(&s_cfe[f * LS + lane * 4]) = v;
            *(v4f*)(out_cfe + ((size_t)b * F_SZ + f) * D_SZ + lane * 4) = v;
        }

        // ---- Kick off next b's prefetch (wave-uniform predicate; results are
        //      consumed only at the next commit, so no wait before the GEMM).
        if (iter + 1 < ITERS) {
            int bn = b + GRID;
#pragma unroll
            for (int r = 0; r < 5; ++r) {
                int f = wave + r * 8;
                cpre[r] = coef[bn * F_SZ + f];
                int ix  = index[bn * F_SZ + f];
                pre[r]  = *(const v4f*)(emb + ((size_t)f * V_SZ + (size_t)ix) * D_SZ
                                            + lane * 4);
            }
        }

        __syncthreads();   // s_cfe visible to all waves

        // ---- GEMM (Ueff @ C_b) via WMMA + fused epilogue reduction.
        float partial = 0.0f;
#pragma unroll
        for (int mt = 0; mt < MT; ++mt) {
            v8f acc = {};
#pragma unroll
            for (int kt = 0; kt < KT; ++kt) {
                // B 4x16 f32 tile: VGPR0: K = half*2, VGPR1: K = half*2 + 1
                int k0 = kt * 4 + half * 2;
                v2f bfrag;
                bfrag.x = s_cfe[(k0 + 0) * LS + dbase + l16];
                bfrag.y = s_cfe[(k0 + 1) * LS + dbase + l16];
                acc = __builtin_amdgcn_wmma_f32_16x16x4_f32(
                        /*neg_a=*/false, afrag[mt * KT + kt],
                        /*neg_b=*/false, bfrag,
                        /*c_mod=*/(short)0, acc,
                        /*reuse_a=*/false, /*reuse_b=*/false);
            }
            // C/D layout: VGPR v -> M = mt*16 + v + (lane<16 ? 0 : 8), N = dbase+l16
#pragma unroll
            for (int v = 0; v < 8; ++v) {
                int f = mt * 16 + v + half * 8;
                float y   = acc[v];
                float cfe = s_cfe[f * LS + dbase + l16];   // 0 for f >= 40
                float fe  = s_fe [f * LS + dbase + l16];   // 0 for f >= 40
                partial += cfe * (fe + y);
            }
        }

        // ---- Reduce: 32 lanes -> 1, then 8 waves -> 1.
#pragma unroll
        for (int off = 16; off > 0; off >>= 1)
            partial += __shfl_xor(partial, off, 32);
        if (lane == 0) s_red[wave] = partial;
        __syncthreads();
        if (tid == 0) {
            float tot = 0.0f;
#pragma unroll
            for (int w = 0; w < 8; ++w) tot += s_red[w];
            out_scalar[b] = w0v + tot;
        }
    }
}

// ---------------------------------------------------------------------------
extern "C" void kernel_launch(void* const* d_in, const int* in_sizes, int n_in,
                              void* d_out, int out_size, void* d_ws, size_t ws_size,
                              hipStream_t stream) {
    const int*   index = (const int*)  d_in[0];
    const float* coef  = (const float*)d_in[1];
    const float* emb   = (const float*)d_in[2];
    const float* fe    = (const float*)d_in[3];
    const float* W     = (const float*)d_in[4];
    const float* w0    = (const float*)d_in[5];

    float* out_scalar = (float*)d_out;           // (B,)
    float* out_cfe    = (float*)d_out + B_SZ;    // (B,F,D)
    float* ws_a       = (float*)d_ws;            // 30*64 floats = 7.5 KB

    fwfm_pack_ueff<<<(MT * KT * 64 + 255) / 256, 256, 0, stream>>>(W, ws_a);
    fwfm_main<<<GRID, 256, 0, stream>>>(index, coef, emb, fe, w0, ws_a,
                                        out_scalar, out_cfe);
}